// LabelProp_6622839570803
// MI455X (gfx1250) — compile-verified
//
#include <hip/hip_runtime.h>
#include <hip/hip_bf16.h>

// LabelProp scatter-mean for MI455X (gfx1250, wave32).
//
// d_in[0] lbls      float32 [N*C]   (C = 32)
// d_in[1] null_mask bool    [N]     (JAX bool = 1 byte/elem -> unsigned char)
// d_in[2] src_sc    int32   [E]
// d_in[3] dst_sc    int32   [E]
// d_in[4] src_fc    int32   [E]
// d_in[5] dst_fc    int32   [E]
// d_out             float32 [N*C]
// d_ws layout (floats): sums_sc[N*C] | sums_fc[N*C] | cnt_sc[N] | cnt_fc[N]
//
// Roofline: FLOPs negligible; lbls (12.8 MB) + accumulators (26.4 MB) fit in
// the 192 MB L2, so the ceiling is L2 atomic-RMW throughput (~106M
// global_atomic_add_f32). Index streams are staged into LDS with gfx1250
// async loads (global_load_async_to_lds_b128 / ASYNCcnt), double-buffered
// against the atomic inner loop. Kept edges are compacted into LDS per wave,
// then processed 4-at-a-time by lane quads with float4 gathers/atomics.

#define NCH   32
#define WAVE  32
#define CHUNK 256                      // edges staged per wave per iteration
#define WAVES_PER_BLOCK 8              // 256 threads

// ---------------------------------------------------------------------------
// Zero the accumulator workspace (must run every launch: atomics accumulate).
// ---------------------------------------------------------------------------
__global__ __launch_bounds__(256) void lp_zero(float4* __restrict__ p, int n4) {
    int i = blockIdx.x * blockDim.x + threadIdx.x;
    if (i < n4) p[i] = make_float4(0.f, 0.f, 0.f, 0.f);
}

// ---------------------------------------------------------------------------
// Stage CHUNK (=256) int32 from global g[ebase..ebase+255] into LDS at `lds`
// with async loads: 32 lanes x 16B = 512B per instruction -> 2 instructions.
// Tracked by ASYNCcnt; caller synchronizes with s_wait_asynccnt.
// Callers invoke only under wave-uniform control (full EXEC).
// ---------------------------------------------------------------------------
__device__ __forceinline__ void stage_chunk_async(int* lds, const int* g,
                                                  int ebase, int lane) {
    // Generic pointer to LDS: low 32 bits are the LDS byte address.
    unsigned l0 = (unsigned)(uintptr_t)lds + (unsigned)lane * 16u;
    unsigned long long gb = (unsigned long long)(uintptr_t)g;
    int v0 = (ebase + lane * 4) * 4;   // byte offset of this lane's 16B
    asm volatile("global_load_async_to_lds_b128 %0, %1, %2"
                 :: "v"(l0), "v"(v0), "s"(gb) : "memory");
    asm volatile("global_load_async_to_lds_b128 %0, %1, %2"
                 :: "v"(l0 + 512u), "v"(v0 + 512), "s"(gb) : "memory");
}

// ---------------------------------------------------------------------------
// Scatter one 32-edge tile:
//  1) ballot kept edges, compact their (src,dst) into per-wave LDS via a
//     popc-prefix (same-wave DS ops retire in order; wave_barrier stops
//     compiler reordering).
//  2) 4 lane-quads each take one compacted edge per iteration; the 8 lanes
//     of a quad are the 8 float4 channel groups: one coalesced 128B
//     global_load_b128 gather of lbls[src][:] and 4 global_atomic_add_f32
//     per lane into sums[dst][:]. 4 edges' gathers in flight per iteration.
// ---------------------------------------------------------------------------
__device__ __forceinline__ void scatter_tile(
    int s, int d, bool active,
    const float4* __restrict__ lbls4,
    float* __restrict__ sums, float* __restrict__ cnt,
    int* __restrict__ cs, int* __restrict__ cd,   // per-wave LDS [32] each
    int lane)
{
    const unsigned m    = (unsigned)__ballot(active);   // wave32: low 32 bits
    const int      nact = __popc(m);
    if (nact == 0) return;

    if (active) {
        const int slot = __popc(m & ((1u << lane) - 1u));
        cs[slot] = s;
        cd[slot] = d;
    }
    __builtin_amdgcn_wave_barrier();          // LDS write -> read ordering
    asm volatile("" ::: "memory");

    const int q  = lane >> 3;                 // quad id: which edge of 4
    const int c4 = lane & 7;                  // float4 channel group

    for (int i = 0; i < nact; i += 4) {
        const int j = i + q;
        if (j < nact) {
            const int dd = cd[j];
            const int ss = cs[j];
            const float4 v = lbls4[ss * (NCH / 4) + c4];
            float* srow = sums + (size_t)dd * NCH + c4 * 4;
            atomicAdd(srow + 0, v.x);
            atomicAdd(srow + 1, v.y);
            atomicAdd(srow + 2, v.z);
            atomicAdd(srow + 3, v.w);
            if (c4 == 0) atomicAdd(cnt + dd, 1.0f);
        }
    }
    __builtin_amdgcn_wave_barrier();          // keep next tile's writes behind
    asm volatile("" ::: "memory");
}

// ---------------------------------------------------------------------------
// Scatter-accumulate one edge list. Grid-stride over 256-edge chunks per
// wave; index streams double-buffered in LDS via async loads.
// ---------------------------------------------------------------------------
__global__ __launch_bounds__(256) void lp_scatter(
    const float* __restrict__ lbls,
    const unsigned char* __restrict__ nmask,
    const int* __restrict__ src,
    const int* __restrict__ dst,
    float* __restrict__ sums,
    float* __restrict__ cnt,
    int E)
{
    // [wave][buf][ src[CHUNK] | dst[CHUNK] ]
    __shared__ int smem[WAVES_PER_BLOCK * 2 * 2 * CHUNK];
    // per-wave compaction scratch: [wave][ cs[32] | cd[32] ]
    __shared__ int comp[WAVES_PER_BLOCK * 2 * WAVE];

    const int lane       = threadIdx.x & (WAVE - 1);
    const int wv         = threadIdx.x >> 5;
    const int totalWaves = gridDim.x * WAVES_PER_BLOCK;
    const int gw         = blockIdx.x * WAVES_PER_BLOCK + wv;
    const int nchunks    = (E + CHUNK - 1) / CHUNK;

    int* wbase = smem + wv * (2 * 2 * CHUNK);
    int* cs    = comp + wv * (2 * WAVE);
    int* cd    = cs + WAVE;

    const float4* lbls4 = (const float4*)lbls;

    // Prologue: stage this wave's first chunk if it is a full chunk.
    if (gw < nchunks && (gw + 1) * CHUNK <= E) {
        stage_chunk_async(wbase,         src, gw * CHUNK, lane);
        stage_chunk_async(wbase + CHUNK, dst, gw * CHUNK, lane);
    }

    int bufsel = 0;
    for (int c = gw; c < nchunks; c += totalWaves) {
        const int  nxt     = c + totalWaves;
        const bool nxtFull = (nxt < nchunks) && ((nxt + 1) * CHUNK <= E);

        if (nxtFull) {
            int* bn = wbase + (bufsel ^ 1) * (2 * CHUNK);
            stage_chunk_async(bn,         src, nxt * CHUNK, lane);
            stage_chunk_async(bn + CHUNK, dst, nxt * CHUNK, lane);
            // 4 newly issued outstanding; <=4 ==> current chunk has landed.
            asm volatile("s_wait_asynccnt 0x4" ::: "memory");
        } else {
            asm volatile("s_wait_asynccnt 0x0" ::: "memory");
        }

        if ((c + 1) * CHUNK <= E) {
            // Full chunk: indices come from LDS (no bounds checks).
            const int* lsrc = wbase + bufsel * (2 * CHUNK);
            const int* ldst = lsrc + CHUNK;
            #pragma unroll
            for (int t = 0; t < CHUNK; t += WAVE) {
                const int d = ldst[t + lane];
                const int s = lsrc[t + lane];
                const bool active = (nmask[d] != 0);
                scatter_tile(s, d, active, lbls4, sums, cnt, cs, cd, lane);
            }
        } else {
            // Tail chunk: direct guarded global loads.
            for (int t = 0; t < CHUNK; t += WAVE) {
                const int e = c * CHUNK + t + lane;
                const bool inb = (e < E);
                const int d = inb ? dst[e] : 0;
                const int s = inb ? src[e] : 0;
                const bool active = inb && (nmask[d] != 0);
                scatter_tile(s, d, active, lbls4, sums, cnt, cs, cd, lane);
            }
        }
        bufsel ^= 1;
    }
}

// ---------------------------------------------------------------------------
// Finalize: out = null ? 0.5*(s1/max(c1,1) + s2/max(c2,1)) : lbls
// float4-vectorized: one thread per 4 channels (8 threads per node).
// ---------------------------------------------------------------------------
__global__ __launch_bounds__(256) void lp_finalize(
    const float4* __restrict__ lbls4,
    const unsigned char* __restrict__ nmask,
    const float4* __restrict__ sums_sc4,
    const float* __restrict__ cnt_sc,
    const float4* __restrict__ sums_fc4,
    const float* __restrict__ cnt_fc,
    float4* __restrict__ out4,
    int n4_total)   // N * (NCH/4)
{
    int i = blockIdx.x * blockDim.x + threadIdx.x;
    if (i >= n4_total) return;
    const int n = i >> 3;   // NCH/4 == 8 float4 per node

    float4 r = lbls4[i];
    if (nmask[n]) {
        const float c1 = fmaxf(cnt_sc[n], 1.0f);
        const float c2 = fmaxf(cnt_fc[n], 1.0f);
        const float4 a = sums_sc4[i];
        const float4 b = sums_fc4[i];
        r.x = (a.x / c1 + b.x / c2) * 0.5f;
        r.y = (a.y / c1 + b.y / c2) * 0.5f;
        r.z = (a.z / c1 + b.z / c2) * 0.5f;
        r.w = (a.w / c1 + b.w / c2) * 0.5f;
    }
    out4[i] = r;
}

// ---------------------------------------------------------------------------
extern "C" void kernel_launch(void* const* d_in, const int* in_sizes, int n_in,
                              void* d_out, int out_size, void* d_ws, size_t ws_size,
                              hipStream_t stream) {
    const float*         lbls   = (const float*)d_in[0];
    const unsigned char* nmask  = (const unsigned char*)d_in[1];
    const int*           src_sc = (const int*)d_in[2];
    const int*           dst_sc = (const int*)d_in[3];
    const int*           src_fc = (const int*)d_in[4];
    const int*           dst_fc = (const int*)d_in[5];
    float*               out    = (float*)d_out;

    const int N  = in_sizes[0] / NCH;   // 100,000
    const int E1 = in_sizes[2];         // 3,200,000
    const int E2 = in_sizes[4];

    float* w       = (float*)d_ws;
    float* sums_sc = w;
    float* sums_fc = w + (size_t)N * NCH;
    float* cnt_sc  = w + (size_t)2 * N * NCH;
    float* cnt_fc  = cnt_sc + N;

    // 1) zero accumulators (float4-wide; counts are divisible by 4)
    const int n4ws = (2 * N * NCH + 2 * N) / 4;
    lp_zero<<<(n4ws + 255) / 256, 256, 0, stream>>>((float4*)d_ws, n4ws);

    // 2) scatter-accumulate both graphs (atomic f32 adds resolve in L2).
    //    1024 blocks x 8 wave32 = 8192 waves over 12500 chunks: most waves
    //    get >=2 chunks so the async double-buffer overlaps fetch + atomics.
    const int scatterBlocks = 1024;
    lp_scatter<<<scatterBlocks, 256, 0, stream>>>(lbls, nmask, src_sc, dst_sc,
                                                  sums_sc, cnt_sc, E1);
    lp_scatter<<<scatterBlocks, 256, 0, stream>>>(lbls, nmask, src_fc, dst_fc,
                                                  sums_fc, cnt_fc, E2);

    // 3) finalize blend
    const int n4 = N * (NCH / 4);
    lp_finalize<<<(n4 + 255) / 256, 256, 0, stream>>>(
        (const float4*)lbls, nmask,
        (const float4*)sums_sc, cnt_sc,
        (const float4*)sums_fc, cnt_fc,
        (float4*)out, n4);
}